// SoftDTW_31894427140829
// MI455X (gfx1250) — compile-verified
//
#include <hip/hip_runtime.h>

#define B_   32
#define C_   80
#define N_   512
#define BIGF 1e30f

typedef float v2f __attribute__((ext_vector_type(2)));
typedef float v8f __attribute__((ext_vector_type(8)));

// ---------------------------------------------------------------------------
// Kernel 1: row norms  xx[b][n] = sum_c x[b][c][n]^2   (and same for yy)
// ---------------------------------------------------------------------------
__global__ void __launch_bounds__(256)
softdtw_norms(const float* __restrict__ x, const float* __restrict__ y,
              float* __restrict__ xx, float* __restrict__ yy) {
    int idx   = blockIdx.x * blockDim.x + threadIdx.x;   // 0 .. 2*B*N-1
    int which = idx / (B_ * N_);
    int r     = idx - which * (B_ * N_);
    int b     = r / N_;
    int n     = r - b * N_;
    const float* src = which ? y : x;
    const float* p   = src + (size_t)b * C_ * N_ + n;
    float s = 0.f;
#pragma unroll 8
    for (int c = 0; c < C_; ++c) {
        float v = p[(size_t)c * N_];
        s += v * v;
    }
    (which ? yy : xx)[r] = s;
}

// ---------------------------------------------------------------------------
// Kernel 2: batched GEMM via V_WMMA_F32_16X16X4_F32
//   D[b][n][m] = xx[b][n] + yy[b][m] - 2 * sum_c x[b][c][n]*y[b][c][m]
// Block: 256 threads = 8 waves, tile 128x128, wave tile 32x64 (2x4 WMMAs).
// ---------------------------------------------------------------------------
__global__ void __launch_bounds__(256)
softdtw_gemm(const float* __restrict__ x, const float* __restrict__ y,
             const float* __restrict__ xx, const float* __restrict__ yy,
             float* __restrict__ D) {
    const int b    = blockIdx.z;
    const int lane = threadIdx.x & 31;
    const int w    = threadIdx.x >> 5;
    const int wr   = w & 3;          // 4 wave-rows  * 32 rows = 128
    const int wc   = w >> 2;         // 2 wave-cols  * 64 cols = 128
    const int n_base = blockIdx.x * 128 + wr * 32;
    const int m_base = blockIdx.y * 128 + wc * 64;
    const int half = lane >> 4;      // 0: K=k0,k0+1   1: K=k0+2,k0+3
    const int l16  = lane & 15;

    const float* xb = x + (size_t)b * C_ * N_;
    const float* yb = y + (size_t)b * C_ * N_;

    v8f acc[2][4] = {};              // 64 VGPRs of f32 accumulation

    for (int k0 = 0; k0 < C_; k0 += 4) {
        const int cA = k0 + 2 * half;
        v2f a[2], bf[4];
        // A-matrix (x^T): 32-bit A 16x4 layout (ISA 7.12.2)
#pragma unroll
        for (int r = 0; r < 2; ++r) {
            int n = n_base + r * 16 + l16;
            a[r][0] = xb[(size_t)cA       * N_ + n];
            a[r][1] = xb[(size_t)(cA + 1) * N_ + n];
        }
        // B-matrix (y): 4x16, column-major load, mirrored layout
#pragma unroll
        for (int c = 0; c < 4; ++c) {
            int m = m_base + c * 16 + l16;
            bf[c][0] = yb[(size_t)cA       * N_ + m];
            bf[c][1] = yb[(size_t)(cA + 1) * N_ + m];
        }
#pragma unroll
        for (int r = 0; r < 2; ++r)
#pragma unroll
            for (int c = 0; c < 4; ++c)
                acc[r][c] = __builtin_amdgcn_wmma_f32_16x16x4_f32(
                    false, a[r], false, bf[c], (short)0, acc[r][c], false, false);
    }

    // Epilogue: C/D 16x16 f32 layout — VGPR p: lanes 0-15 -> M=p, lanes 16-31 -> M=p+8
#pragma unroll
    for (int c = 0; c < 4; ++c) {
        int   m  = m_base + c * 16 + l16;
        float yv = yy[b * N_ + m];
#pragma unroll
        for (int r = 0; r < 2; ++r) {
#pragma unroll
            for (int p = 0; p < 8; ++p) {
                int   n  = n_base + r * 16 + p + 8 * half;
                float xv = xx[b * N_ + n];
                D[((size_t)b * N_ + n) * N_ + m] = xv + yv - 2.0f * acc[r][c][p];
            }
        }
    }
}

// ---------------------------------------------------------------------------
// Kernel 3: soft-DTW wavefront, one block (512 threads) per batch.
// 4 rotating LDS diagonals -> single barrier per anti-diagonal step.
// ---------------------------------------------------------------------------
__global__ void __launch_bounds__(512)
softdtw_dp(const float* __restrict__ D, float* __restrict__ out) {
    __shared__ float buf[4][N_ + 1];
    const int b   = blockIdx.x;
    const int tid = threadIdx.x;
    const int i   = tid + 1;                 // this thread owns row i (1..512)

    // d=0 diagonal in buf[0] (R[0][0]=0, rest BIG); d=1 diagonal in buf[1] (all BIG)
    buf[0][i] = BIGF;
    buf[1][i] = BIGF;
    if (tid == 0) { buf[0][0] = 0.0f; buf[1][0] = BIGF; }
    __syncthreads();

    const float* Db = D + (size_t)b * N_ * N_;

    for (int d = 2; d <= 2 * N_; ++d) {
        float*       cur = buf[d & 3];
        const float* p1  = buf[(d - 1) & 3];
        const float* p2  = buf[(d - 2) & 3];
        int   j = d - i;
        float v = BIGF;
        if (j >= 1 && j <= N_) {
            float r0 = p2[i - 1];            // R[i-1][j-1]
            float r1 = p1[i - 1];            // R[i-1][j]
            float r2 = p1[i];                // R[i][j-1]
            float mn = fminf(r0, fminf(r1, r2));
            float s  = __expf(mn - r0) + __expf(mn - r1) + __expf(mn - r2);
            // gamma = 1:  softmin = mn - log(sum exp(mn - r_k))
            v = Db[(size_t)(i - 1) * N_ + (j - 1)] + (mn - __logf(s));
        }
        cur[i] = v;
        if (tid == 0) cur[0] = BIGF;
        __syncthreads();
    }
    if (tid == 0) out[b] = buf[(2 * N_) & 3][N_];   // R[N][M]
}

// ---------------------------------------------------------------------------
extern "C" void kernel_launch(void* const* d_in, const int* in_sizes, int n_in,
                              void* d_out, int out_size, void* d_ws, size_t ws_size,
                              hipStream_t stream) {
    (void)in_sizes; (void)n_in; (void)out_size; (void)ws_size;
    const float* x = (const float*)d_in[0];
    const float* y = (const float*)d_in[1];

    float* D  = (float*)d_ws;                       // 32*512*512 f32 = 33.55 MB
    float* xx = D + (size_t)B_ * N_ * N_;           // 32*512 f32
    float* yy = xx + B_ * N_;                       // 32*512 f32

    softdtw_norms<<<(2 * B_ * N_) / 256, 256, 0, stream>>>(x, y, xx, yy);
    softdtw_gemm<<<dim3(4, 4, B_), 256, 0, stream>>>(x, y, xx, yy, D);
    softdtw_dp<<<B_, 512, 0, stream>>>(D, (float*)d_out);
}